// GCNEncoder_52853867545095
// MI455X (gfx1250) — compile-verified
//
#include <hip/hip_runtime.h>
#include <math.h>

// ---------------------------------------------------------------------------
// GCN encoder for MI455X (gfx1250, wave32).
//   h   = relu( Agg( x @ W1 ) + b1 )      Agg = self-loop + deg^-1/2 sym norm
//   out =       Agg( h @ W2 ) + b2
// GEMMs: V_WMMA_F32_16X16X4_F32 (full fp32 precision, LDS-staged A strips,
// register-resident W fragments). Edge aggregation: one wave per edge,
// float4/float2 coalesced gathers, hardware global_atomic_add_f32 scatters.
// ---------------------------------------------------------------------------

typedef float v2f __attribute__((ext_vector_type(2)));
typedef float v8f __attribute__((ext_vector_type(8)));

// Guaranteed hardware fp32 atomic add (no return -> STOREcnt; drained by the
// implicit wait-idle at s_endpgm).
static __device__ __forceinline__ void atomicAddF32(float* p, float v) {
    asm volatile("global_atomic_add_f32 %0, %1, off" :: "v"(p), "v"(v) : "memory");
}

// ---------------------------- degree / dinv --------------------------------

__global__ void zero_deg_kernel(int* __restrict__ deg, int n) {
    int i = blockIdx.x * blockDim.x + threadIdx.x;
    if (i < n) deg[i] = 0;
}

__global__ void count_deg_kernel(const int* __restrict__ dst, int* __restrict__ deg, int e) {
    int i = blockIdx.x * blockDim.x + threadIdx.x;
    if (i < e) atomicAdd(&deg[dst[i]], 1);
}

__global__ void dinv_kernel(const int* __restrict__ deg, float* __restrict__ dinv, int n) {
    int i = blockIdx.x * blockDim.x + threadIdx.x;
    if (i < n) dinv[i] = rsqrtf((float)deg[i] + 1.0f);   // +1 = self loop
}

// ------------------------------ WMMA GEMM ----------------------------------
// C[nrows, NOUT] = A'[nrows, 128] @ W[128, NOUT]
// A' = RELU_BIAS ? relu(A + b1[col]) : A
// Block: NOUT/16 waves; each wave owns a 16-col tile, W frags in registers.

template <int NOUT, bool RELU_BIAS>
__global__ void gemm_wmma_kernel(const float* __restrict__ A,
                                 const float* __restrict__ W,
                                 const float* __restrict__ bias,
                                 float* __restrict__ C,
                                 int nrows) {
    constexpr int K = 128;
    constexpr int NWAVES = NOUT / 16;
    __shared__ float sA[16 * K];                         // 8 KB

    const int lane = threadIdx.x & 31;
    const int wave = threadIdx.x >> 5;
    const int n0   = wave * 16;
    const int ncol = n0 + (lane & 15);
    const int koff = (lane & 16) ? 2 : 0;

    v2f wf[32];
#pragma unroll
    for (int kb = 0; kb < 32; ++kb) {
        const int k = kb * 4 + koff;
        wf[kb].x = W[(size_t)k * NOUT + ncol];
        wf[kb].y = W[(size_t)(k + 1) * NOUT + ncol];
    }

    const int m    = lane & 15;
    const int mrow = (lane & 16) ? 8 : 0;

    for (int row0 = blockIdx.x * 16; row0 + 16 <= nrows; row0 += gridDim.x * 16) {
        const float4* gsrc = (const float4*)(A + (size_t)row0 * K);
        for (int idx4 = threadIdx.x; idx4 < (16 * K) / 4; idx4 += NWAVES * 32) {
            float4 v = gsrc[idx4];
            if constexpr (RELU_BIAS) {
                const int k = (idx4 * 4) & (K - 1);
                v.x = fmaxf(v.x + bias[k + 0], 0.0f);
                v.y = fmaxf(v.y + bias[k + 1], 0.0f);
                v.z = fmaxf(v.z + bias[k + 2], 0.0f);
                v.w = fmaxf(v.w + bias[k + 3], 0.0f);
            }
            ((float4*)sA)[idx4] = v;
        }
        __syncthreads();

        v8f c = {0.f, 0.f, 0.f, 0.f, 0.f, 0.f, 0.f, 0.f};
#pragma unroll
        for (int kb = 0; kb < 32; ++kb) {
            const v2f a = *(const v2f*)&sA[m * K + kb * 4 + koff];
            c = __builtin_amdgcn_wmma_f32_16x16x4_f32(
                    false, a, false, wf[kb], (short)0, c, false, false);
        }

        float* cp = C + (size_t)(row0 + mrow) * NOUT + n0 + m;
#pragma unroll
        for (int g = 0; g < 8; ++g) cp[(size_t)g * NOUT] = c[g];

        __syncthreads();
    }
}

// ----------------------- self-loop / bias init passes ----------------------
// acc[i,f] = dinv[i]^2 * hl[i,f]                (layer 1, 128 feats, float4)
__global__ void self_init_kernel(const float* __restrict__ hl,
                                 const float* __restrict__ dinv,
                                 float* __restrict__ acc, int total4) {
    int idx4 = blockIdx.x * blockDim.x + threadIdx.x;
    if (idx4 < total4) {
        const int i = idx4 >> 5;                 // 128/4 float4 per row
        const float d = dinv[i];
        const float w = d * d;
        float4 v = ((const float4*)hl)[idx4];
        v.x *= w; v.y *= w; v.z *= w; v.w *= w;
        ((float4*)acc)[idx4] = v;
    }
}

// out[i,f] = dinv[i]^2 * hl2[i,f] + b2[f]       (layer 2, 64 feats, float4)
__global__ void out_init_kernel(const float* __restrict__ hl2,
                                const float* __restrict__ dinv,
                                const float* __restrict__ b2,
                                float* __restrict__ out, int total4) {
    int idx4 = blockIdx.x * blockDim.x + threadIdx.x;
    if (idx4 < total4) {
        const int i  = idx4 >> 4;                // 64/4 float4 per row
        const int f4 = (idx4 & 15);
        const float d = dinv[i];
        const float w = d * d;
        float4 v = ((const float4*)hl2)[idx4];
        float4 b = ((const float4*)b2)[f4];
        v.x = v.x * w + b.x; v.y = v.y * w + b.y;
        v.z = v.z * w + b.z; v.w = v.w * w + b.w;
        ((float4*)out)[idx4] = v;
    }
}

// ------------------------------ edge scatter -------------------------------
// One wave32 per edge. Each lane gathers FEAT/32 contiguous floats of
// hl[src] (vector load) and issues hardware fp32 atomics into acc[dst].

template <int FEAT>
__global__ void scatter_kernel(const float* __restrict__ hl,
                               const float* __restrict__ dinv,
                               const int* __restrict__ src,
                               const int* __restrict__ dst,
                               float* __restrict__ acc, int e) {
    constexpr int VEC = FEAT / 32;               // floats per lane (4 or 2)
    const int lane = threadIdx.x & 31;
    const int eidx = blockIdx.x * (blockDim.x >> 5) + (threadIdx.x >> 5);
    if (eidx >= e) return;
    const int s = src[eidx];
    const int d = dst[eidx];
    const float w = dinv[s] * dinv[d];
    const float* gp = hl  + (size_t)s * FEAT + lane * VEC;
    float*       ap = acc + (size_t)d * FEAT + lane * VEC;
    if constexpr (VEC == 4) {
        float4 v = *(const float4*)gp;
        atomicAddF32(ap + 0, w * v.x);
        atomicAddF32(ap + 1, w * v.y);
        atomicAddF32(ap + 2, w * v.z);
        atomicAddF32(ap + 3, w * v.w);
    } else {
        float2 v = *(const float2*)gp;
        atomicAddF32(ap + 0, w * v.x);
        atomicAddF32(ap + 1, w * v.y);
    }
}

// ------------------------------- launcher ----------------------------------

extern "C" void kernel_launch(void* const* d_in, const int* in_sizes, int n_in,
                              void* d_out, int out_size, void* d_ws, size_t ws_size,
                              hipStream_t stream) {
    const float* x   = (const float*)d_in[0];
    const int*   ei  = (const int*)d_in[1];
    const float* W1  = (const float*)d_in[2];
    const float* b1  = (const float*)d_in[3];
    const float* W2  = (const float*)d_in[4];
    const float* b2  = (const float*)d_in[5];
    float*       out = (float*)d_out;

    const int N = in_sizes[0] / 128;
    const int E = in_sizes[1] / 2;
    const int* src = ei;
    const int* dst = ei + E;

    // Workspace layout (256B aligned slices).
    char* w = (char*)d_ws;
    auto carve = [&](size_t bytes) {
        void* p = (void*)w;
        w += (bytes + 255) & ~(size_t)255;
        return p;
    };
    int*   deg  = (int*)carve((size_t)N * 4);
    float* dinv = (float*)carve((size_t)N * 4);
    float* hl1  = (float*)carve((size_t)N * 128 * 4);   // reused as hl2
    float* hagg = (float*)carve((size_t)N * 128 * 4);
    float* hl2  = hl1;
    (void)n_in; (void)out_size; (void)ws_size;

    const int T = 256;
    // 1) degrees + dinv
    zero_deg_kernel<<<(N + T - 1) / T, T, 0, stream>>>(deg, N);
    count_deg_kernel<<<(E + T - 1) / T, T, 0, stream>>>(dst, deg, E);
    dinv_kernel<<<(N + T - 1) / T, T, 0, stream>>>(deg, dinv, N);

    // 2) hl1 = x @ W1   (WMMA, 8 waves = 8 column tiles)
    gemm_wmma_kernel<128, false><<<2048, 256, 0, stream>>>(x, W1, nullptr, hl1, N);

    // 3) layer-1 aggregation: self-loop init then edge atomics (1 wave/edge)
    {
        const int total4 = N * 128 / 4;
        self_init_kernel<<<(total4 + T - 1) / T, T, 0, stream>>>(hl1, dinv, hagg, total4);
        scatter_kernel<128><<<(E + 7) / 8, 256, 0, stream>>>(hl1, dinv, src, dst, hagg, E);
    }

    // 4) hl2 = relu(hagg + b1) @ W2   (bias+relu fused into A-load)
    gemm_wmma_kernel<64, true><<<2048, 128, 0, stream>>>(hagg, W2, b1, hl2, N);

    // 5) layer-2 aggregation into d_out, b2 folded into init
    {
        const int total4 = N * 64 / 4;
        out_init_kernel<<<(total4 + T - 1) / T, T, 0, stream>>>(hl2, dinv, b2, out, total4);
        scatter_kernel<64><<<(E + 7) / 8, 256, 0, stream>>>(hl2, dinv, src, dst, out, E);
    }
}